// QuadtreeAttention_42855183679493
// MI455X (gfx1250) — compile-verified
//
#include <hip/hip_runtime.h>
#include <hip/hip_bf16.h>
#include <math.h>

typedef __attribute__((ext_vector_type(2))) float v2f;
typedef __attribute__((ext_vector_type(8))) float v8f;

#define B_   2
#define N_   1600
#define C_   256
#define NH   8
#define HD   32
#define TOPK 8
#define H_   40
#define W_   40
#define WC   20
#define K1_  400
#define SF   0.17677669529663687f   /* 32^-0.5 */
#define OUTB (N_*N_ + N_*K1_)       /* per-batch output rows: 3,200,000 */

// ---------------------------------------------------------------------------
// Kernel 1: q = x @ Wq^T and k = target @ Wk^T via V_WMMA_F32_16X16X4_F32.
// One wave computes one 16x16 output tile; 64 WMMA k-steps (K=256).
// A layout (ISA 7.12.2): lane L holds A[m=L&15][k0 + (L>>4)*2 + {0,1}]
// B layout (mirrored):   lane L holds B[k0 + (L>>4)*2 + {0,1}][n=L&15]
// C/D layout: lane L, vgpr i -> D[m = i + (L>>4)*8][n = L&15]
// ---------------------------------------------------------------------------
__global__ void __launch_bounds__(128)
qk_proj_kernel(const float* __restrict__ x, const float* __restrict__ tgt,
               const float* __restrict__ Wq, const float* __restrict__ Wk,
               float* __restrict__ qb, float* __restrict__ kb)
{
    const int wave = threadIdx.x >> 5;
    const int lane = threadIdx.x & 31;
    const int tile = blockIdx.x * 4 + wave;          // 0..6399
    const int mat  = (tile >= 3200) ? 1 : 0;
    const int t    = mat ? (tile - 3200) : tile;
    const int mt   = t >> 4;                          // 0..199 (rows of B*N=3200)
    const int nt   = t & 15;                          // 0..15  (cols of C=256)

    const float* A = mat ? tgt : x;
    const float* W = mat ? Wk  : Wq;
    float*       O = mat ? kb  : qb;

    const int half = lane >> 4;
    const int l16  = lane & 15;
    const float* ap = A + (size_t)(mt * 16 + l16) * C_ + half * 2;
    const float* bp = W + (size_t)(nt * 16 + l16) * C_ + half * 2;   // Wq[n][k]

    v8f acc = {};
    for (int k0 = 0; k0 < C_; k0 += 4) {
        v2f a, b;
        a.x = ap[k0]; a.y = ap[k0 + 1];
        b.x = bp[k0]; b.y = bp[k0 + 1];
        acc = __builtin_amdgcn_wmma_f32_16x16x4_f32(false, a, false, b,
                                                    (short)0, acc, false, false);
    }
    float* op = O + (size_t)(mt * 16 + half * 8) * C_ + nt * 16 + l16;
#pragma unroll
    for (int i = 0; i < 8; ++i) op[(size_t)i * C_] = acc[i];
}

// ---------------------------------------------------------------------------
// Kernel 2: coarse keys = 2x2 avg pool of k, stored (B, nh, K1, hd) so the
// s1 GEMM's B-operand reads contiguous float2s.
// ---------------------------------------------------------------------------
__global__ void pool_kernel(const float* __restrict__ kb, float* __restrict__ k1b)
{
    const int i = blockIdx.x * blockDim.x + threadIdx.x;   // B*C*K1 = 204800
    if (i >= B_ * C_ * K1_) return;
    const int d = i & 31;
    const int j = (i >> 5) % K1_;
    const int h = (i / (32 * K1_)) & 7;
    const int b = i / (32 * K1_ * NH);
    const int r = j / WC, cc = j % WC;
    const int c = h * HD + d;
    const int n0 = (2 * r) * W_ + 2 * cc;
    const float* base = kb + (size_t)b * N_ * C_ + c;
    float v = base[(size_t)n0 * C_] + base[(size_t)(n0 + 1) * C_] +
              base[(size_t)(n0 + W_) * C_] + base[(size_t)(n0 + W_ + 1) * C_];
    k1b[i] = 0.25f * v;
}

// ---------------------------------------------------------------------------
// Kernel 3: coarse scores S[b,h,q,j] = sf * <q_h[q,:], k1[b,h,j,:]>.
// Per (b,h): (1600x32) x (32x400) GEMM; one wave per 16x16 tile, 8 WMMA steps.
// ---------------------------------------------------------------------------
__global__ void __launch_bounds__(128)
s1_kernel(const float* __restrict__ qb, const float* __restrict__ k1b,
          float* __restrict__ Sb)
{
    const int wave = threadIdx.x >> 5;
    const int lane = threadIdx.x & 31;
    int tile = blockIdx.x * 4 + wave;       // 0..39999
    const int nt = tile % 25; tile /= 25;   // 25 n-tiles (K1=400)
    const int mt = tile % 100; tile /= 100; // 100 m-tiles (N=1600)
    const int bh = tile;                    // 0..15
    const int b  = bh >> 3, h = bh & 7;

    const int half = lane >> 4;
    const int l16  = lane & 15;
    const float* ap = qb  + (size_t)(b * N_ + mt * 16 + l16) * C_ + h * HD + half * 2;
    const float* bp = k1b + (size_t)(bh * K1_ + nt * 16 + l16) * HD + half * 2;

    v8f acc = {};
#pragma unroll
    for (int k0 = 0; k0 < HD; k0 += 4) {
        v2f a, bb;
        a.x  = ap[k0]; a.y  = ap[k0 + 1];
        bb.x = bp[k0]; bb.y = bp[k0 + 1];
        acc = __builtin_amdgcn_wmma_f32_16x16x4_f32(false, a, false, bb,
                                                    (short)0, acc, false, false);
    }
    float* op = Sb + ((size_t)bh * N_ + mt * 16 + half * 8) * K1_ + nt * 16 + l16;
#pragma unroll
    for (int i = 0; i < 8; ++i) op[(size_t)i * K1_] = acc[i] * SF;
}

// ---------------------------------------------------------------------------
// Kernel 4: fused per-(b,q) block (8 waves = 8 heads):
//   softmax(400) -> top-8 -> fine candidate softmax(32) * parent prob
//   -> deterministic sparse scatter into LDS out0 row -> head_proj outputs.
// ---------------------------------------------------------------------------
__global__ void __launch_bounds__(256)
quadtree_kernel(const float* __restrict__ qb, const float* __restrict__ kb,
                const float* __restrict__ Sb, const float* __restrict__ Whp,
                const float* __restrict__ bhp, float* __restrict__ out)
{
    __shared__ float attR[NH][K1_];     // normalized coarse attention rows
    __shared__ float row0[N_];          // dense fine output row accumulator
    __shared__ float whps[NH];
    __shared__ float bsh;
    __shared__ int   scatIdx[256];
    __shared__ float scatVal[256];

    const int b    = blockIdx.x / N_;
    const int qi   = blockIdx.x % N_;
    const int tid  = threadIdx.x;
    const int h    = tid >> 5;          // wave == head
    const int lane = tid & 31;

    if (tid < NH) whps[tid] = Whp[tid];
    if (tid == 0) bsh = bhp[0];
    for (int j = tid; j < N_; j += 256) row0[j] = 0.0f;
    __syncthreads();

    // ---- coarse softmax over 400 scores (13 regs per lane) ----
    const float* srow = Sb + ((size_t)(b * NH + h) * N_ + qi) * K1_;
    const float NEG = -__builtin_inff();
    float v[13];
#pragma unroll
    for (int t = 0; t < 13; ++t) {
        const int j = lane + t * 32;
        v[t] = (j < K1_) ? srow[j] : NEG;
    }
    float m = NEG;
#pragma unroll
    for (int t = 0; t < 13; ++t) m = fmaxf(m, v[t]);
#pragma unroll
    for (int off = 16; off > 0; off >>= 1) m = fmaxf(m, __shfl_xor(m, off, 32));
    float s = 0.0f;
#pragma unroll
    for (int t = 0; t < 13; ++t) {
        const int j = lane + t * 32;
        v[t] = (j < K1_) ? __expf(v[t] - m) : 0.0f;
        s += v[t];
    }
#pragma unroll
    for (int off = 16; off > 0; off >>= 1) s += __shfl_xor(s, off, 32);
    const float inv = 1.0f / s;
#pragma unroll
    for (int t = 0; t < 13; ++t) {
        const int j = lane + t * 32;
        if (j < K1_) { v[t] *= inv; attR[h][j] = v[t]; }
        else         { v[t] = -1.0f; }                 // below any probability
    }

    // ---- top-8 by iterative wave argmax (smaller index wins ties) ----
    float pv[TOPK]; int pidx[TOPK];
#pragma unroll
    for (int it = 0; it < TOPK; ++it) {
        float bv = -2.0f; int bj = 0;
#pragma unroll
        for (int t = 0; t < 13; ++t) {
            const int j = lane + t * 32;
            if (v[t] > bv) { bv = v[t]; bj = j; }
        }
#pragma unroll
        for (int off = 16; off > 0; off >>= 1) {
            const float ov = __shfl_xor(bv, off, 32);
            const int   oj = __shfl_xor(bj, off, 32);
            if (ov > bv || (ov == bv && oj < bj)) { bv = ov; bj = oj; }
        }
        pv[it] = bv; pidx[it] = bj;
        if ((bj & 31) == lane) {
            const int tt = bj >> 5;
#pragma unroll
            for (int t = 0; t < 13; ++t) if (t == tt) v[t] = -1.0f;
        }
    }

    // ---- fine stage: lane = one of 32 candidates (4 children x 8 parents) --
    const int t4    = lane >> 2;
    const int child = lane & 3;
    float pvsel = 0.0f; int parent = 0;
#pragma unroll
    for (int it = 0; it < TOPK; ++it)
        if (it == t4) { pvsel = pv[it]; parent = pidx[it]; }

    const int r  = parent / WC, cc = parent % WC;
    const int n  = (2 * r) * W_ + 2 * cc + ((child & 2) ? W_ : 0) + (child & 1);
    const float* qp = qb + (size_t)(b * N_ + qi) * C_ + h * HD;
    const float* kp = kb + (size_t)(b * N_ + n)  * C_ + h * HD;
    float dot = 0.0f;
#pragma unroll
    for (int d = 0; d < HD; ++d) dot += qp[d] * kp[d];
    dot *= SF;

    float fm = dot;
#pragma unroll
    for (int off = 16; off > 0; off >>= 1) fm = fmaxf(fm, __shfl_xor(fm, off, 32));
    const float fe = __expf(dot - fm);
    float fs = fe;
#pragma unroll
    for (int off = 16; off > 0; off >>= 1) fs += __shfl_xor(fs, off, 32);
    const float p0 = (fe / fs) * pvsel;

    // deterministic scatter: record, then one lane replays in fixed order
    scatIdx[tid] = n;
    scatVal[tid] = p0 * whps[h];
    __syncthreads();
    if (tid == 0) {
        for (int i = 0; i < 256; ++i) row0[scatIdx[i]] += scatVal[i];
    }
    __syncthreads();

    // ---- outputs: [out0 (N*N) | out1 (N*K1)] per batch ----
    const float bb  = bsh;
    const size_t ob = (size_t)b * OUTB;
    for (int j = tid; j < K1_; j += 256) {
        float acc = bb;
#pragma unroll
        for (int hh = 0; hh < NH; ++hh) acc += attR[hh][j] * whps[hh];
        out[ob + (size_t)N_ * N_ + (size_t)qi * K1_ + j] = acc;
    }
    for (int j = tid; j < N_; j += 256) {
        out[ob + (size_t)qi * N_ + j] = row0[j] + bb;
    }
}

// ---------------------------------------------------------------------------
extern "C" void kernel_launch(void* const* d_in, const int* in_sizes, int n_in,
                              void* d_out, int out_size, void* d_ws, size_t ws_size,
                              hipStream_t stream)
{
    const float* x    = (const float*)d_in[0];
    const float* tgt  = (const float*)d_in[1];
    const float* Wq   = (const float*)d_in[2];
    const float* Wk   = (const float*)d_in[3];
    const float* Whp  = (const float*)d_in[4];
    const float* bhp  = (const float*)d_in[5];
    float* out = (float*)d_out;

    float* qb  = (float*)d_ws;                       // B*N*C   =   819,200 f32
    float* kb  = qb  + (size_t)B_ * N_ * C_;         // B*N*C   =   819,200 f32
    float* k1b = kb  + (size_t)B_ * N_ * C_;         // B*C*K1  =   204,800 f32
    float* Sb  = k1b + (size_t)B_ * C_ * K1_;        // B*nh*N*K1 = 10,240,000 f32

    // 1) q/k projections: 6400 wave-tiles / 4 waves per block
    qk_proj_kernel<<<1600, 128, 0, stream>>>(x, tgt, Wq, Wk, qb, kb);
    // 2) 2x2 average pool -> coarse keys
    pool_kernel<<<(B_ * C_ * K1_ + 255) / 256, 256, 0, stream>>>(kb, k1b);
    // 3) coarse scores: 40000 wave-tiles / 4 waves per block
    s1_kernel<<<10000, 128, 0, stream>>>(qb, k1b, Sb);
    // 4) fused softmax / top-k / fine attention / projection
    quadtree_kernel<<<B_ * N_, 256, 0, stream>>>(qb, kb, Sb, Whp, bhp, out);
}